// PINNSimulator_26474178413198
// MI455X (gfx1250) — compile-verified
//
#include <hip/hip_runtime.h>
#include <hip/hip_bf16.h>
#include <stdint.h>

// ---------------------------------------------------------------------------
// Problem constants: shape (2, 1, 192, 192, 192) fp32, 20000 samples/batch
// ---------------------------------------------------------------------------
#define DIM        192
#define PLANE      (192 * 192)          // 36864
#define VOL        (192 * 192 * 192)    // 7077888 per batch
#define BATCH      2
#define NTOT       (BATCH * VOL)        // 14155776
#define NPTS       20000
#define NBINS      2048
#define BINS_PER_THREAD 8               // 2048 / 256

typedef float v2f __attribute__((ext_vector_type(2)));
typedef float v8f __attribute__((ext_vector_type(8)));

// Workspace layout (bytes)
//   0    : float sum_ic
//   4    : float sum_phys
//   8    : uint  cnt_hi[2]
//   16   : uint  cnt_tie[2]
//   24   : uint  n_hi[2]
//   32   : uint  thresh_bin[2]
//   64   : uint  hist[2][NBINS]              (16 KB)
//   16448: int   idx[2][NPTS]                (160 KB)
#define WS_SUM_IC(ws)    ((float*)((char*)(ws) + 0))
#define WS_SUM_PHYS(ws)  ((float*)((char*)(ws) + 4))
#define WS_CNT_HI(ws)    ((unsigned*)((char*)(ws) + 8))
#define WS_CNT_TIE(ws)   ((unsigned*)((char*)(ws) + 16))
#define WS_N_HI(ws)      ((unsigned*)((char*)(ws) + 24))
#define WS_THRESH(ws)    ((unsigned*)((char*)(ws) + 32))
#define WS_HIST(ws)      ((unsigned*)((char*)(ws) + 64))
#define WS_IDX(ws)       ((int*)((char*)(ws) + 16448))
#define WS_ZERO_BYTES    16448

// ---------------------------------------------------------------------------
// xor-exchange within the 32-lane wave via DS_SWIZZLE_B32 (group-of-32 mode:
// offset[4:0]=and_mask=0x1f, offset[9:5]=or_mask=0, offset[14:10]=xor_mask).
// Pattern is a template non-type parameter so the builtin sees an ICE.
// ---------------------------------------------------------------------------
template <int XORM>
__device__ __forceinline__ float swz_add(float x) {
    constexpr int pattern = (XORM << 10) | 0x1f;
    int y = __builtin_amdgcn_ds_swizzle(__float_as_int(x), pattern);
    return x + __int_as_float(y);
}

// ---------------------------------------------------------------------------
// Wave reduction through the matrix pipe.
// A = ones(16x4)  =>  D[m][n] = sum_k B[k][n] : every D row is the vector of
// column sums of B. Each lane contributes one B element (b.x = partial,
// b.y = 0); the WMMA collapses the 4-deep K groups in one matrix-pipe op,
// then 4 ds_swizzle xor steps fold the 16 column sums.
// EXEC must be all ones at the call site (callers guarantee convergence).
// ---------------------------------------------------------------------------
__device__ __forceinline__ void wave_accum_wmma(float v, float* gsum) {
    v2f a;  a[0] = 1.0f; a[1] = 1.0f;     // ones A-matrix (16x4 striped)
    v2f bm; bm[0] = v;   bm[1] = 0.0f;    // lane partial into B (4x16 striped)
    v8f c = {};
    c = __builtin_amdgcn_wmma_f32_16x16x4_f32(
            /*neg_a=*/false, a, /*neg_b=*/false, bm,
            /*c_mod=*/(short)0, c, /*reuse_a=*/false, /*reuse_b=*/false);
    float s = c[0];                       // column sum for N = lane%16
    s = swz_add<1>(s);
    s = swz_add<2>(s);
    s = swz_add<4>(s);
    s = swz_add<8>(s);                    // each 16-group now holds the total
    if ((threadIdx.x & 31u) == 0u) atomicAdd(gsum, s);
}

// ---------------------------------------------------------------------------
// Deterministic hash-Gumbel (stand-in for jax.random.gumbel(key=42)):
// key_i = log(u_i + 1e-4) + G(hash(b, i))
// ---------------------------------------------------------------------------
__device__ __forceinline__ uint32_t pcg_hash(uint32_t x) {
    x = x * 747796405u + 2891336453u;
    uint32_t w = ((x >> ((x >> 28u) + 4u)) ^ x) * 277803737u;
    return (w >> 22u) ^ w;
}

__device__ __forceinline__ float samp_key(float uval, uint32_t b, uint32_t i) {
    uint32_t h = pcg_hash(i ^ (b * 0x9E3779B9u) ^ 0x85EBCA6Bu);
    float un = (float)(h >> 8) * (1.0f / 16777216.0f);
    un = fminf(fmaxf(un, 1e-7f), 1.0f - 1e-7f);
    float g = -__logf(-__logf(un));
    return __logf(uval + 1e-4f) + g;
}

// monotone float -> uint mapping for descending radix threshold
__device__ __forceinline__ uint32_t key_bits(float k) {
    uint32_t x = __float_as_uint(k);
    return (x & 0x80000000u) ? ~x : (x | 0x80000000u);
}

// ---------------------------------------------------------------------------
// Kernel 1: IC loss partial sums  (float4 streaming, WGP-scope prefetch,
// WMMA reduce). Prefetch index is clamped (cndmask) instead of branched.
// ---------------------------------------------------------------------------
__global__ void ic_loss_kernel(const float* __restrict__ u0,
                               const float* __restrict__ t0,
                               float* __restrict__ ws_sum) {
    const float4* a4 = (const float4*)u0;
    const float4* b4 = (const float4*)t0;
    const int n4 = NTOT / 4;
    const int stride = gridDim.x * blockDim.x;
    float acc = 0.0f;
    for (int i = blockIdx.x * blockDim.x + threadIdx.x; i < n4; i += stride) {
        int pi = i + stride;
        pi = (pi < n4) ? pi : i;                     // clamp, no branch
        __builtin_prefetch(&a4[pi], 0, 3);           // WGP-scope prefetch
        __builtin_prefetch(&b4[pi], 0, 3);
        float4 a = a4[i];
        float4 b = b4[i];
        float d0 = a.x - b.x, d1 = a.y - b.y, d2 = a.z - b.z, d3 = a.w - b.w;
        acc += d0 * d0 + d1 * d1 + d2 * d2 + d3 * d3;
    }
    wave_accum_wmma(acc, ws_sum);
}

// ---------------------------------------------------------------------------
// Kernel 2: histogram of sampling keys (LDS-privatized, 2 batches x 2048 bins)
// ---------------------------------------------------------------------------
__global__ void key_hist_kernel(const float* __restrict__ u,
                                unsigned* __restrict__ ghist) {
    __shared__ unsigned lhist[BATCH * NBINS];
    for (int i = threadIdx.x; i < BATCH * NBINS; i += blockDim.x) lhist[i] = 0u;
    __syncthreads();

    const float4* u4 = (const float4*)u;
    const int n4 = NTOT / 4;
    const int stride = gridDim.x * blockDim.x;
    for (int i = blockIdx.x * blockDim.x + threadIdx.x; i < n4; i += stride) {
        int pi = i + stride;
        pi = (pi < n4) ? pi : i;
        __builtin_prefetch(&u4[pi], 0, 3);
        float4 v = u4[i];
        uint32_t base = (uint32_t)i * 4u;
        uint32_t b = base / (uint32_t)VOL;     // batch (VOL % 4 == 0)
        uint32_t loc = base - b * (uint32_t)VOL;
        float vv[4] = {v.x, v.y, v.z, v.w};
        #pragma unroll
        for (int k = 0; k < 4; ++k) {
            uint32_t bin = key_bits(samp_key(vv[k], b, loc + k)) >> 21;
            atomicAdd(&lhist[b * NBINS + bin], 1u);
        }
    }
    __syncthreads();
    for (int i = threadIdx.x; i < BATCH * NBINS; i += blockDim.x) {
        unsigned c = lhist[i];
        if (c) atomicAdd(&ghist[i], c);
    }
}

// ---------------------------------------------------------------------------
// Kernel 3: parallel suffix scan over bins -> threshold bin & above-count.
// One block per batch; 256 threads x 8 bins each (descending ownership),
// prefix over the 256 partials in LDS, exact bin localized by its owner.
// ---------------------------------------------------------------------------
__global__ void scan_kernel(const unsigned* __restrict__ ghist,
                            unsigned* __restrict__ n_hi,
                            unsigned* __restrict__ thresh) {
    __shared__ unsigned part[256];
    __shared__ unsigned pref[256];
    int b = blockIdx.x;
    int t = threadIdx.x;
    // thread t owns bins [hi-7 .. hi], hi descending with t
    int hi = NBINS - 1 - t * BINS_PER_THREAD;
    unsigned local = 0;
    #pragma unroll
    for (int k = 0; k < BINS_PER_THREAD; ++k)
        local += ghist[b * NBINS + hi - k];
    part[t] = local;
    __syncthreads();
    if (t == 0) {                       // cheap serial prefix over 256 in LDS
        unsigned acc = 0;
        for (int j = 0; j < 256; ++j) { pref[j] = acc; acc += part[j]; }
    }
    __syncthreads();
    unsigned before = pref[t];          // count in bins strictly above my range
    if (before < (unsigned)NPTS && before + part[t] >= (unsigned)NPTS) {
        unsigned acc = before;
        #pragma unroll
        for (int k = 0; k < BINS_PER_THREAD; ++k) {
            unsigned c = ghist[b * NBINS + hi - k];
            if (acc + c >= (unsigned)NPTS) {
                thresh[b] = (unsigned)(hi - k);
                n_hi[b] = acc;
                break;
            }
            acc += c;
        }
    }
}

// ---------------------------------------------------------------------------
// Kernel 4: select indices. bin > T -> definite top-k; bin == T -> first
// (NPTS - n_hi) ties fill the remainder (bounded atomic counter).
// ---------------------------------------------------------------------------
__global__ void select_kernel(const float* __restrict__ u,
                              const unsigned* __restrict__ n_hi,
                              const unsigned* __restrict__ thresh,
                              unsigned* __restrict__ cnt_hi,
                              unsigned* __restrict__ cnt_tie,
                              int* __restrict__ idx) {
    const int stride = gridDim.x * blockDim.x;
    for (int i = blockIdx.x * blockDim.x + threadIdx.x; i < NTOT; i += stride) {
        uint32_t b = (uint32_t)i / (uint32_t)VOL;
        uint32_t loc = (uint32_t)i - b * (uint32_t)VOL;
        uint32_t bin = key_bits(samp_key(u[i], b, loc)) >> 21;
        unsigned T = thresh[b];
        if (bin > T) {
            unsigned pos = atomicAdd(&cnt_hi[b], 1u);
            idx[b * NPTS + pos] = (int)loc;
        } else if (bin == T) {
            unsigned hi = n_hi[b];
            unsigned rem = (unsigned)NPTS - hi;
            unsigned pos = atomicAdd(&cnt_tie[b], 1u);
            if (pos < rem) idx[b * NPTS + hi + pos] = (int)loc;
        }
    }
}

// ---------------------------------------------------------------------------
// Kernel 5: PDE residual at sampled points only. The 7-point Laplacian is
// evaluated on-the-fly per point (7 gathers) — dense conv never materialized.
// ---------------------------------------------------------------------------
__global__ void physics_kernel(const float* __restrict__ u,
                               const float* __restrict__ dudt,
                               const float* __restrict__ rawD,
                               const float* __restrict__ rawR,
                               const int* __restrict__ idx,
                               float* __restrict__ ws_sum) {
    int t = blockIdx.x * blockDim.x + threadIdx.x;
    float sq = 0.0f;
    if (t < BATCH * NPTS) {
        int b = t / NPTS;
        int loc = idx[t];
        const float* ub = u + (size_t)b * VOL;
        int x = loc % DIM;
        int y = (loc / DIM) % DIM;
        int z = loc / PLANE;
        float uc = ub[loc];
        float lap = -6.0f * uc;
        lap += (x > 0)        ? ub[loc - 1]     : 0.0f;
        lap += (x < DIM - 1)  ? ub[loc + 1]     : 0.0f;
        lap += (y > 0)        ? ub[loc - DIM]   : 0.0f;
        lap += (y < DIM - 1)  ? ub[loc + DIM]   : 0.0f;
        lap += (z > 0)        ? ub[loc - PLANE] : 0.0f;
        lap += (z < DIM - 1)  ? ub[loc + PLANE] : 0.0f;

        size_t g = (size_t)b * VOL + (size_t)loc;
        float sD = 1.0f / (1.0f + __expf(-rawD[g]));
        float sR = 1.0f / (1.0f + __expf(-rawR[g]));
        float D   = 0.001f + sD * (0.02f  - 0.001f);
        float rho = 0.012f + sR * (0.034f - 0.012f);
        float r = dudt[g] - D * lap - rho * uc * (1.0f - uc);
        sq = r * r;
    }
    wave_accum_wmma(sq, ws_sum);   // all lanes converged here; EXEC all ones
}

// ---------------------------------------------------------------------------
// Kernel 6: finalize the three scalars
// ---------------------------------------------------------------------------
__global__ void finalize_kernel(const float* __restrict__ sum_ic,
                                const float* __restrict__ sum_phys,
                                float* __restrict__ out) {
    if (threadIdx.x == 0 && blockIdx.x == 0) {
        float ic = sum_ic[0] * (1.0f / (float)NTOT);
        float ph = sum_phys[0] * (1.0f / (float)(BATCH * NPTS));
        out[0] = 10.0f * ic + ph;
        out[1] = ic;
        out[2] = ph;
    }
}

// ---------------------------------------------------------------------------
extern "C" void kernel_launch(void* const* d_in, const int* in_sizes, int n_in,
                              void* d_out, int out_size, void* d_ws, size_t ws_size,
                              hipStream_t stream) {
    const float* u_pred_t  = (const float*)d_in[0];
    const float* u_pred_t0 = (const float*)d_in[1];
    const float* target_t0 = (const float*)d_in[2];
    const float* du_dt_map = (const float*)d_in[3];
    const float* raw_D     = (const float*)d_in[4];
    const float* raw_rho   = (const float*)d_in[5];
    float* out = (float*)d_out;

    // Zero accumulators / counters / histograms (graph-capture safe).
    (void)hipMemsetAsync(d_ws, 0, WS_ZERO_BYTES, stream);

    const int TPB = 256;
    const int STREAM_BLOCKS = 2048;   // fills the WGPs for the streaming passes

    ic_loss_kernel<<<STREAM_BLOCKS, TPB, 0, stream>>>(
        u_pred_t0, target_t0, WS_SUM_IC(d_ws));

    key_hist_kernel<<<STREAM_BLOCKS, TPB, 0, stream>>>(
        u_pred_t, WS_HIST(d_ws));

    scan_kernel<<<BATCH, 256, 0, stream>>>(
        WS_HIST(d_ws), WS_N_HI(d_ws), WS_THRESH(d_ws));

    select_kernel<<<STREAM_BLOCKS, TPB, 0, stream>>>(
        u_pred_t, WS_N_HI(d_ws), WS_THRESH(d_ws),
        WS_CNT_HI(d_ws), WS_CNT_TIE(d_ws), WS_IDX(d_ws));

    int phys_blocks = (BATCH * NPTS + TPB - 1) / TPB;
    physics_kernel<<<phys_blocks, TPB, 0, stream>>>(
        u_pred_t, du_dt_map, raw_D, raw_rho, WS_IDX(d_ws), WS_SUM_PHYS(d_ws));

    finalize_kernel<<<1, 32, 0, stream>>>(
        WS_SUM_IC(d_ws), WS_SUM_PHYS(d_ws), out);
}